// MasifLigandNet_12506944766307
// MI455X (gfx1250) — compile-verified
//
#include <hip/hip_runtime.h>
#include <hip/hip_bf16.h>

// MasifLigandNet head for MI455X (gfx1250, wave32).
// Pipeline: zero-mask -> per-(b,l) wave32 top-10 -> deterministic masked mean
// -> fp32 WMMA GEMM (V_WMMA_F32_16X16X4_F32) -> BN+SiLU -> tiny GEMM2.

#define B_ 64
#define N_ 4096
#define L_ 64
#define D_ 512
#define O_ 7
#define KSEL 10

typedef float v2f __attribute__((ext_vector_type(2)));
typedef float v8f __attribute__((ext_vector_type(8)));

// ---------------------------------------------------------------- zero mask
__global__ __launch_bounds__(256) void k_zero_mask(float* __restrict__ mask) {
    int t = blockIdx.x * 256 + threadIdx.x;
    if (t < B_ * N_) mask[t] = 0.0f;
}

// ------------------------------------------------- top-K selection (1 wave
// per (b, l) column). Squared distance preserves top-k ordering (skip sqrt).
// Each lane keeps a sorted 10-entry register list over its 128 candidates,
// then 10 rounds of wave-wide lexicographic min-merge via shuffles.
__global__ __launch_bounds__(32) void k_topk(const float* __restrict__ pos,
                                             const float* __restrict__ lig,
                                             float* __restrict__ mask) {
    const int bl = blockIdx.x;          // 0 .. B*L-1
    const int b  = bl / L_;
    const int l  = bl % L_;
    const int lane = threadIdx.x;

    const float lx = lig[(b * L_ + l) * 3 + 0];
    const float ly = lig[(b * L_ + l) * 3 + 1];
    const float lz = lig[(b * L_ + l) * 3 + 2];

    float    bd[KSEL];
    unsigned bi[KSEL];
#pragma unroll
    for (int j = 0; j < KSEL; ++j) { bd[j] = __builtin_inff(); bi[j] = 0xFFFFFFFFu; }

    const float* pb = pos + (size_t)b * N_ * 3;
    for (int i = lane; i < N_; i += 32) {
        float dx = pb[i * 3 + 0] - lx;
        float dy = pb[i * 3 + 1] - ly;
        float dz = pb[i * 3 + 2] - lz;
        float d2 = dx * dx + dy * dy + dz * dz;
        if (d2 < bd[KSEL - 1]) {            // bubble-insert, constant indices only
            float cd = d2; unsigned ci = (unsigned)i;
#pragma unroll
            for (int j = 0; j < KSEL; ++j) {
                bool sm = (cd < bd[j]) || (cd == bd[j] && ci < bi[j]);
                float td = bd[j]; unsigned ti = bi[j];
                if (sm) { bd[j] = cd; bi[j] = ci; cd = td; ci = ti; }
            }
        }
    }

    float* mb = mask + (size_t)b * N_;
    for (int r = 0; r < KSEL; ++r) {
        float d = bd[0]; unsigned idx = bi[0];
#pragma unroll
        for (int off = 16; off > 0; off >>= 1) {   // wave32 butterfly min
            float    od = __shfl_xor(d, off);
            unsigned oi = (unsigned)__shfl_xor((int)idx, off);
            if (od < d || (od == d && oi < idx)) { d = od; idx = oi; }
        }
        if (bd[0] == d && bi[0] == idx) {          // unique winner pops its head
#pragma unroll
            for (int j = 0; j < KSEL - 1; ++j) { bd[j] = bd[j + 1]; bi[j] = bi[j + 1]; }
            bd[KSEL - 1] = __builtin_inff(); bi[KSEL - 1] = 0xFFFFFFFFu;
        }
        if (lane == 0) mb[idx] = 1.0f;             // scatter-set dedups repeats
    }
}

// ------------------------------------------------- masked mean of x rows.
// Deterministic ballot/popc compaction into LDS (ascending index order ->
// bit-stable float accumulation on every graph replay), then each of 512
// threads owns one feature dim and gathers only the <=640 selected rows.
__global__ __launch_bounds__(512) void k_emb(const float* __restrict__ x,
                                             const float* __restrict__ mask,
                                             float* __restrict__ emb) {
    __shared__ int s_idx[L_ * KSEL];   // max 640 unique rows
    __shared__ int s_cnt[16];
    __shared__ int s_base[17];

    const int b    = blockIdx.x;
    const int tid  = threadIdx.x;
    const int wave = tid >> 5;
    const int lane = tid & 31;
    const float* mb = mask + (size_t)b * N_;
    const int rows = N_ / 16;          // 256 rows per wave
    const int r0   = wave * rows;

    int cnt = 0;                       // pass 1: per-wave count (ballot-uniform)
    for (int i = r0 + lane; i < r0 + rows; i += 32) {
        unsigned long long bal = __ballot(mb[i] != 0.0f);
        cnt += __popcll(bal);
    }
    if (lane == 0) s_cnt[wave] = cnt;
    __syncthreads();
    if (tid == 0) {
        int acc = 0;
        for (int w = 0; w < 16; ++w) { s_base[w] = acc; acc += s_cnt[w]; }
        s_base[16] = acc;
    }
    __syncthreads();

    int off = s_base[wave];            // pass 2: deterministic compaction
    const unsigned long long ltmask = ((unsigned long long)1 << lane) - 1ull;
    for (int i = r0 + lane; i < r0 + rows; i += 32) {
        bool sel = mb[i] != 0.0f;
        unsigned long long bal = __ballot(sel);
        if (sel) s_idx[off + __popcll(bal & ltmask)] = i;
        off += __popcll(bal);
    }
    __syncthreads();

    const int   total = s_base[16];
    const float inv   = 1.0f / (float)total;
    const float* xb   = x + (size_t)b * N_ * D_;
    float acc = 0.0f;                  // pass 3: gather+sum in ascending order
    for (int p = 0; p < total; ++p) {
        if (p + 8 < total)
            __builtin_prefetch(&xb[(size_t)s_idx[p + 8] * D_ + tid], 0, 0);
        acc += xb[(size_t)s_idx[p] * D_ + tid];
    }
    emb[(size_t)b * D_ + tid] = acc * inv;
}

// ------------------------------------------------- h = emb @ W1^T + b1 via
// V_WMMA_F32_16X16X4_F32. One wave per 16x16 output tile, K stepped by 4.
// A(16x4) layout: lanes 0-15 hold K={0,1}, lanes 16-31 hold K={2,3}.
// B(4x16)=W1^T tile mirrors A addressing since B[k][n] = W1[n][k].
__global__ __launch_bounds__(32) void k_gemm1(const float* __restrict__ emb,
                                              const float* __restrict__ W1,
                                              const float* __restrict__ b1,
                                              float* __restrict__ h) {
    const int n0   = blockIdx.x * 16;      // output feature tile
    const int m0   = blockIdx.y * 16;      // batch tile
    const int lane = threadIdx.x;
    const int half = lane >> 4;
    const int l16  = lane & 15;

    const float* arow = emb + (size_t)(m0 + l16) * D_ + 2 * half;
    const float* brow = W1  + (size_t)(n0 + l16) * D_ + 2 * half;

    v8f acc = {};
    for (int k = 0; k < D_; k += 4) {
        v2f a, bb;
        a.x  = arow[k + 0];  a.y  = arow[k + 1];   // 8B contiguous -> b64 load
        bb.x = brow[k + 0];  bb.y = brow[k + 1];
        acc = __builtin_amdgcn_wmma_f32_16x16x4_f32(
            false, a, false, bb, (short)0, acc, false, false);
    }
#pragma unroll
    for (int r = 0; r < 8; ++r) {          // C/D: VGPR r <-> M = r (+8 for hi half)
        int row = m0 + r + 8 * half;
        int col = n0 + l16;
        h[(size_t)row * D_ + col] = acc[r] + b1[col];
    }
}

// ------------------------------------------------- batchnorm (training stats,
// biased var over B=64) + SiLU, in place. One thread per feature.
__global__ __launch_bounds__(512) void k_bn_silu(float* __restrict__ h,
                                                 const float* __restrict__ gamma,
                                                 const float* __restrict__ beta) {
    const int j = threadIdx.x;
    float s = 0.0f, s2 = 0.0f;
    for (int b = 0; b < B_; ++b) {
        float v = h[(size_t)b * D_ + j];
        s += v; s2 += v * v;
    }
    const float mu   = s * (1.0f / B_);
    const float var  = s2 * (1.0f / B_) - mu * mu;
    const float rstd = rsqrtf(var + 1e-5f);
    const float g = gamma[j], be = beta[j];
    for (int b = 0; b < B_; ++b) {
        float v = h[(size_t)b * D_ + j];
        v = g * (v - mu) * rstd + be;
        v = v / (1.0f + expf(-v));         // SiLU
        h[(size_t)b * D_ + j] = v;
    }
}

// ------------------------------------------------- out = h @ W2^T + b2  [64,7]
__global__ __launch_bounds__(256) void k_gemm2(const float* __restrict__ h,
                                               const float* __restrict__ W2,
                                               const float* __restrict__ b2,
                                               float* __restrict__ out) {
    int t = blockIdx.x * 256 + threadIdx.x;
    if (t >= B_ * O_) return;
    int b = t / O_, o = t % O_;
    const float* hr = h  + (size_t)b * D_;
    const float* wr = W2 + (size_t)o * D_;
    float acc = 0.0f;
    for (int k = 0; k < D_; ++k) acc += hr[k] * wr[k];
    out[t] = acc + b2[o];
}

// ---------------------------------------------------------------- launcher
extern "C" void kernel_launch(void* const* d_in, const int* in_sizes, int n_in,
                              void* d_out, int out_size, void* d_ws, size_t ws_size,
                              hipStream_t stream) {
    const float* pos   = (const float*)d_in[0];
    const float* x     = (const float*)d_in[1];
    const float* lig   = (const float*)d_in[2];
    const float* W1    = (const float*)d_in[3];
    const float* b1    = (const float*)d_in[4];
    const float* gamma = (const float*)d_in[5];
    const float* beta  = (const float*)d_in[6];
    const float* W2    = (const float*)d_in[7];
    const float* b2    = (const float*)d_in[8];

    float* ws   = (float*)d_ws;
    float* mask = ws;                        // B*N   = 262144 floats
    float* emb  = mask + (size_t)B_ * N_;    // B*D   =  32768 floats
    float* h    = emb  + (size_t)B_ * D_;    // B*D   =  32768 floats

    k_zero_mask<<<(B_ * N_ + 255) / 256, 256, 0, stream>>>(mask);
    k_topk<<<B_ * L_, 32, 0, stream>>>(pos, lig, mask);
    k_emb<<<B_, 512, 0, stream>>>(x, mask, emb);
    k_gemm1<<<dim3(D_ / 16, B_ / 16), 32, 0, stream>>>(emb, W1, b1, h);
    k_bn_silu<<<1, 512, 0, stream>>>(h, gamma, beta);
    k_gemm2<<<(B_ * O_ + 255) / 256, 256, 0, stream>>>(h, W2, b2, (float*)d_out);
}